// ConvexTriplane_69140383531571
// MI455X (gfx1250) — compile-verified
//
#include <hip/hip_runtime.h>
#include <stdint.h>

typedef uint32_t u32;
typedef __attribute__((ext_vector_type(4))) u32 tdm_g0_t;  // D# group 0 (128b)
typedef __attribute__((ext_vector_type(8))) int  tdm_g1_t; // D# group 1 (256b)
typedef __attribute__((ext_vector_type(4))) int  tdm_g2_t; // D# groups 2/3 (128b)
typedef __attribute__((ext_vector_type(8))) int  tdm_g4_t; // extra 256b arg (clang-23 form)

// Low 32 bits of a flat pointer into __shared__ == LDS byte offset
// (flat->LDS mapping truncates to addr[31:0], ISA 10.2).
__device__ __forceinline__ u32 lds_offset(const void* p) {
  return (u32)(uintptr_t)p;
}

// TDM: DMA rows 63..64 (2 floats) of a (64ch x 128) line table for all 64
// channels into LDS, packed as 64 x {float lo, float hi}.
// Tensor: 2D, dim0=128 (N, stride 128), dim1=64 (C). Tile: 2 x 64, tile start
// at element 63 of dim0. 512 bytes moved per descriptor.
__device__ __forceinline__ void tdm_load_line_pair(const float* table, u32 lds_byte_off) {
  uint64_t ga = (uint64_t)(uintptr_t)table + 63ull * 4ull;  // tile start
  tdm_g0_t g0;
  g0.x = 1u;                                   // count=1 (valid user descriptor)
  g0.y = lds_byte_off;                         // lds_addr [63:32]
  g0.z = (u32)ga;                              // global_addr[31:0]
  g0.w = ((u32)(ga >> 32) & 0x01FFFFFFu) | (2u << 30);  // global_addr[56:32] | type=2
  tdm_g1_t g1;
  g1[0] = (int)(2u << 16);                     // wg_mask=0, data_size=2 (4B)
  g1[1] = (int)(128u << 16);                   // tensor_dim0 = 128  (bits 63:48)
  g1[2] = (int)(64u  << 16);                   // tensor_dim1 = 64   (bits 95:80)
  g1[3] = (int)(2u   << 16);                   // tile_dim0   = 2    (bits 127:112)
  g1[4] = 64;                                  // tile_dim1   = 64   (bits 143:128)
  g1[5] = 128;                                 // tensor_dim0_stride = 128 (bits 191:160)
  g1[6] = (int)((8192u & 0xFFFFu) << 16);      // tensor_dim1_stride[15:0] (bits 223:208)
  g1[7] = 0;                                   // tensor_dim1_stride[47:16]
  tdm_g2_t gz = {0, 0, 0, 0};                  // dims 2..4 unused (tile dims = 0)
  tdm_g4_t gz8 = {0, 0, 0, 0, 0, 0, 0, 0};
  // amdgpu-toolchain clang-23 form: (g0, g1, g2, g3, <256b>, cpol).
  __builtin_amdgcn_tensor_load_to_lds(g0, g1, gz, gz, gz8, 0);
}

template<int C, int N>
__device__ __forceinline__ float bilinear_gated(const float* __restrict__ E,
                                                const float* __restrict__ G,
                                                float gx, float gy) {
  const float sc  = 0.5f * (float)(N - 1);
  const float ix  = (gx + 1.0f) * sc;
  const float iy  = (gy + 1.0f) * sc;
  const float x0f = floorf(ix), y0f = floorf(iy);
  const float wx  = ix - x0f,   wy  = iy - y0f;
  const int x0 = (int)x0f, y0 = (int)y0f;
  const int x1 = x0 + 1,   y1 = y0 + 1;
  const float vx0 = (x0 >= 0 && x0 < N) ? 1.0f : 0.0f;
  const float vx1 = (x1 >= 0 && x1 < N) ? 1.0f : 0.0f;
  const float vy0 = (y0 >= 0 && y0 < N) ? 1.0f : 0.0f;
  const float vy1 = (y1 >= 0 && y1 < N) ? 1.0f : 0.0f;
  const int x0c = min(max(x0, 0), N - 1), x1c = min(max(x1, 0), N - 1);
  const int y0c = min(max(y0, 0), N - 1), y1c = min(max(y1, 0), N - 1);
  const float w00 = (1.0f - wx) * (1.0f - wy) * vx0 * vy0;
  const float w01 = wx          * (1.0f - wy) * vx1 * vy0;
  const float w10 = (1.0f - wx) * wy          * vx0 * vy1;
  const float w11 = wx          * wy          * vx1 * vy1;
  const int b00 = y0c * N + x0c, b01 = y0c * N + x1c;
  const int b10 = y1c * N + x0c, b11 = y1c * N + x1c;
  float acc = 0.0f;
#pragma unroll 4
  for (int c = 0; c < C; ++c) {
    const float* e = E + c * (N * N);
    const float* g = G + c * (N * N);
    const float ev = w00 * e[b00] + w01 * e[b01] + w10 * e[b10] + w11 * e[b11];
    const float gv = w00 * g[b00] + w01 * g[b01] + w10 * g[b10] + w11 * g[b11];
    acc += (gv >= 0.0f) ? ev : 0.0f;
  }
  return acc;
}

template<int C, int N>
__device__ __forceinline__ float trilinear_gated(const float* __restrict__ E,
                                                 const float* __restrict__ G,
                                                 float gx, float gy, float gz) {
  const float sc = 0.5f * (float)(N - 1);
  const float ix = (gx + 1.0f) * sc;
  const float iy = (gy + 1.0f) * sc;
  const float iz = (gz + 1.0f) * sc;
  const float x0f = floorf(ix), y0f = floorf(iy), z0f = floorf(iz);
  const float wx = ix - x0f, wy = iy - y0f, wz = iz - z0f;
  const int x0 = (int)x0f, y0 = (int)y0f, z0 = (int)z0f;
  int   xc[2], yc[2], zc[2];
  float wxv[2], wyv[2], wzv[2];
#pragma unroll
  for (int k = 0; k < 2; ++k) {
    const int xs = x0 + k, ys = y0 + k, zs = z0 + k;
    wxv[k] = (k ? wx : 1.0f - wx) * ((xs >= 0 && xs < N) ? 1.0f : 0.0f);
    wyv[k] = (k ? wy : 1.0f - wy) * ((ys >= 0 && ys < N) ? 1.0f : 0.0f);
    wzv[k] = (k ? wz : 1.0f - wz) * ((zs >= 0 && zs < N) ? 1.0f : 0.0f);
    xc[k] = min(max(xs, 0), N - 1);
    yc[k] = min(max(ys, 0), N - 1);
    zc[k] = min(max(zs, 0), N - 1);
  }
  int   off[8];
  float w[8];
#pragma unroll
  for (int dz = 0; dz < 2; ++dz)
#pragma unroll
    for (int dy = 0; dy < 2; ++dy)
#pragma unroll
      for (int dx = 0; dx < 2; ++dx) {
        const int t = (dz << 2) | (dy << 1) | dx;
        off[t] = (zc[dz] * N + yc[dy]) * N + xc[dx];
        w[t]   = wxv[dx] * wyv[dy] * wzv[dz];
      }
  float acc = 0.0f;
#pragma unroll 4
  for (int c = 0; c < C; ++c) {
    const float* e = E + c * (N * N * N);
    const float* g = G + c * (N * N * N);
    float ev = 0.0f, gv = 0.0f;
#pragma unroll
    for (int t = 0; t < 8; ++t) {
      ev += w[t] * e[off[t]];
      gv += w[t] * g[off[t]];
    }
    acc += (gv >= 0.0f) ? ev : 0.0f;
  }
  return acc;
}

__global__ __launch_bounds__(256)
void ConvexTriplane_69140383531571_kernel(
    const float* __restrict__ coords,
    const float* __restrict__ pxy,  const float* __restrict__ pyz,  const float* __restrict__ pxz,
    const float* __restrict__ pxyg, const float* __restrict__ pyzg, const float* __restrict__ pxzg,
    const float* __restrict__ lx,   const float* __restrict__ ly,   const float* __restrict__ lz,
    const float* __restrict__ lxg,  const float* __restrict__ lyg,  const float* __restrict__ lzg,
    const float* __restrict__ vol,  const float* __restrict__ volg,
    float* __restrict__ out, int P)
{
  __shared__ float s_lines[6 * 64 * 2];  // 6 tables x 64 ch x {row63,row64}
  __shared__ float s_lf[64];
  __shared__ float s_linefeat;

  const int tid = threadIdx.x;

  // Wave 0 stages line-table rows 63/64 into LDS via the Tensor Data Mover.
  // Wave-uniform branch so waves 1..7 skip the (EXEC-agnostic) TDM issue.
  if ((tid >> 5) == 0) {
    const u32 base = lds_offset(s_lines);
    tdm_load_line_pair(lx,  base + 0 * 512);
    tdm_load_line_pair(ly,  base + 1 * 512);
    tdm_load_line_pair(lz,  base + 2 * 512);
    tdm_load_line_pair(lxg, base + 3 * 512);
    tdm_load_line_pair(lyg, base + 4 * 512);
    tdm_load_line_pair(lzg, base + 5 * 512);
    __builtin_amdgcn_s_wait_tensorcnt((short)0);
  }
  __syncthreads();

  // Per-channel gated line terms (W==1 => ix=0, iy=63.5 constant => scalar).
  if (tid < 64) {
    const int c = tid;
    float lf = 0.0f;
#pragma unroll
    for (int a = 0; a < 3; ++a) {
      const float e = 0.5f * (s_lines[a * 128 + c * 2] + s_lines[a * 128 + c * 2 + 1]);
      const float g = 0.5f * (s_lines[(a + 3) * 128 + c * 2] + s_lines[(a + 3) * 128 + c * 2 + 1]);
      lf += (g >= 0.0f) ? e : 0.0f;
    }
    s_lf[c] = lf;
  }
  __syncthreads();
  if (tid == 0) {  // deterministic fixed-order reduction
    float s = 0.0f;
    for (int i = 0; i < 64; ++i) s += s_lf[i];
    s_linefeat = s;
  }
  __syncthreads();

  const int p = blockIdx.x * 256 + tid;
  if (p >= P) return;

  // Warm L2 with the 32 MB of volume tables (whole working set is L2-resident).
  const u32 vmask = 16u * 64u * 64u * 64u - 1u;  // 2^22 - 1 elements
  __builtin_prefetch(vol  + (((u32)p * 32u) & vmask), 0, 1);
  __builtin_prefetch(volg + (((u32)p * 32u) & vmask), 0, 1);

  const float x = coords[p * 3 + 0];
  const float y = coords[p * 3 + 1];
  const float z = coords[p * 3 + 2];

  float acc = s_linefeat;
  acc += bilinear_gated<32, 128>(pxy, pxyg, x, y);
  acc += bilinear_gated<32, 128>(pyz, pyzg, y, z);
  acc += bilinear_gated<32, 128>(pxz, pxzg, x, z);
  acc += trilinear_gated<16, 64>(vol, volg, x, y, z);
  out[p] = acc;
}

extern "C" void kernel_launch(void* const* d_in, const int* in_sizes, int n_in,
                              void* d_out, int out_size, void* d_ws, size_t ws_size,
                              hipStream_t stream) {
  const float* coords = (const float*)d_in[0];
  const float* pxy    = (const float*)d_in[1];
  const float* pyz    = (const float*)d_in[2];
  const float* pxz    = (const float*)d_in[3];
  const float* pxyg   = (const float*)d_in[4];
  const float* pyzg   = (const float*)d_in[5];
  const float* pxzg   = (const float*)d_in[6];
  const float* lx     = (const float*)d_in[7];
  const float* ly     = (const float*)d_in[8];
  const float* lz     = (const float*)d_in[9];
  const float* lxg    = (const float*)d_in[10];
  const float* lyg    = (const float*)d_in[11];
  const float* lzg    = (const float*)d_in[12];
  const float* vol    = (const float*)d_in[13];
  const float* volg   = (const float*)d_in[14];
  float* out = (float*)d_out;

  const int P = in_sizes[0] / 3;
  const int blocks = (P + 255) / 256;
  ConvexTriplane_69140383531571_kernel<<<blocks, 256, 0, stream>>>(
      coords, pxy, pyz, pxz, pxyg, pyzg, pxzg,
      lx, ly, lz, lxg, lyg, lzg, vol, volg, out, P);
}